// C_TABLE_ALL_25202868092937
// MI455X (gfx1250) — compile-verified
//
#include <hip/hip_runtime.h>

// Problem constants (reference: b=4, N=2048, K=8, fp32)
#define NDIM 2048
#define KSTEPS 8
#define BATCH 4
#define TPB 256
#define EPT (NDIM / TPB)   // 8 elements per thread
#define POS_BIG_F 1e30f

// gfx1250 async global->LDS copy path (ASYNCcnt), with portable fallback.
#if defined(__HIP_DEVICE_COMPILE__) && defined(__gfx1250__) && \
    __has_builtin(__builtin_amdgcn_global_load_async_to_lds_b128)
#define USE_ASYNC_LDS 1
#define AS1 __attribute__((address_space(1)))
#define AS3 __attribute__((address_space(3)))
typedef int v4i __attribute__((ext_vector_type(4)));
typedef AS1 v4i* async_gptr;   // global (AS1) int4*  -- builtin param 0
typedef AS3 v4i* async_lptr;   // LDS    (AS3) int4*  -- builtin param 1
#else
#define USE_ASYNC_LDS 0
#endif

// k = 0: C0 = D[:, :, N-1]; Call0 row = -1 everywhere except col N-1 = +1.
__global__ __launch_bounds__(TPB) void ctab_step0(
    const float* __restrict__ D, float* __restrict__ C,
    float* __restrict__ Call, float* __restrict__ Cbuf0)
{
  const int bn  = blockIdx.x;         // b*N + n
  const int tid = threadIdx.x;
  float* outRow = Call + ((size_t)bn * KSTEPS + 0) * NDIM;
#pragma unroll
  for (int j = 0; j < EPT; ++j) {
    const int i = tid + j * TPB;
    __builtin_nontemporal_store((i == NDIM - 1) ? 1.0f : -1.0f, outRow + i);
  }
  if (tid == 0) {
    const float c0 = D[(size_t)bn * NDIM + (NDIM - 1)];
    C[bn * KSTEPS + 0] = c0;   // C[:, :, 0]
    Cbuf0[bn]          = c0;   // DP carry for k = 1
  }
}

// k >= 1: per row n of batch b:
//   M[i]  = D[b,n,i] + Cprev[b,i+1]   for n <= i <= N-1-k
//   Ck    = min_i M[i]                (0 for invalid rows n > N-1-k)
//   Call  = softmax_i(-M)             (-1 outside mask / invalid rows)
__global__ __launch_bounds__(TPB) void ctab_step(
    const float* __restrict__ D, const float* __restrict__ Cprev,
    float* __restrict__ Ccur, float* __restrict__ C,
    float* __restrict__ Call, const int kk)
{
  const int bn  = blockIdx.x;
  const int b   = bn >> 11;            // NDIM == 2048
  const int n   = bn & (NDIM - 1);
  const int tid = threadIdx.x;

  float* outRow = Call + ((size_t)bn * KSTEPS + kk) * NDIM;
  const int hi  = NDIM - 1 - kk;       // last active column

  if (n > hi) {                        // invalid row (uniform per block)
#pragma unroll
    for (int j = 0; j < EPT; ++j)
      __builtin_nontemporal_store(-1.0f, outRow + tid + j * TPB);
    if (tid == 0) {
      C[bn * KSTEPS + kk] = 0.0f;
      Ccur[bn]            = 0.0f;
    }
    return;
  }

  __shared__ float ldsD[NDIM];         // staged D row (8 KB)
  __shared__ float redmin[TPB / 32];
  __shared__ float redsum[TPB / 32];

  const float* Drow = D + (size_t)bn * NDIM;
  const float* Cp   = Cprev + (size_t)b * NDIM;   // read Cp[i+1]

  // Stage only the needed span [n & ~3, N) of the D row into LDS.
  const int v0 = (n & ~3) >> 2;        // first float4 index (16B aligned)
#if USE_ASYNC_LDS
  for (int v = v0 + tid; v < NDIM / 4; v += TPB) {
    __builtin_amdgcn_global_load_async_to_lds_b128(
        (async_gptr)(Drow + 4 * v), (async_lptr)(ldsD + 4 * v), 0, 0);
  }
#if __has_builtin(__builtin_amdgcn_s_wait_asynccnt)
  __builtin_amdgcn_s_wait_asynccnt(0);
#else
  asm volatile("s_wait_asynccnt 0x0" ::: "memory");
#endif
  __syncthreads();
#else
  for (int v = v0 + tid; v < NDIM / 4; v += TPB) {
    const float4 t = *(const float4*)(Drow + 4 * v);
    *(float4*)(ldsD + 4 * v) = t;
  }
  __syncthreads();
#endif

  // Pass 1: M values in registers, block-wide min (wave32 shuffles).
  float mv[EPT];
  float lmin = POS_BIG_F;
#pragma unroll
  for (int j = 0; j < EPT; ++j) {
    const int i = tid + j * TPB;
    float M = POS_BIG_F;
    if (i >= n && i <= hi) M = ldsD[i] + Cp[i + 1];
    mv[j] = M;
    lmin = fminf(lmin, M);
  }
#pragma unroll
  for (int off = 16; off > 0; off >>= 1)
    lmin = fminf(lmin, __shfl_xor(lmin, off, 32));
  if ((tid & 31) == 0) redmin[tid >> 5] = lmin;
  __syncthreads();
  float gmin = redmin[0];
#pragma unroll
  for (int w = 1; w < TPB / 32; ++w) gmin = fminf(gmin, redmin[w]);

  // Pass 2: e = exp(-(M - min M)) = exp(gmin - M); block-wide sum.
  float lsum = 0.0f;
#pragma unroll
  for (int j = 0; j < EPT; ++j) {
    const int i = tid + j * TPB;
    float e = 0.0f;
    if (i >= n && i <= hi) e = expf(gmin - mv[j]);
    mv[j] = e;
    lsum += e;
  }
#pragma unroll
  for (int off = 16; off > 0; off >>= 1)
    lsum += __shfl_xor(lsum, off, 32);
  if ((tid & 31) == 0) redsum[tid >> 5] = lsum;
  __syncthreads();
  float gsum = redsum[0];
#pragma unroll
  for (int w = 1; w < TPB / 32; ++w) gsum += redsum[w];
  const float inv = 1.0f / gsum;

  // Pass 3: normalized softmax / -1 fill, streamed past L2 (NT stores).
#pragma unroll
  for (int j = 0; j < EPT; ++j) {
    const int i = tid + j * TPB;
    const float o = (i >= n && i <= hi) ? mv[j] * inv : -1.0f;
    __builtin_nontemporal_store(o, outRow + i);
  }
  if (tid == 0) {
    C[bn * KSTEPS + kk] = gmin;   // row min
    Ccur[bn]            = gmin;   // DP carry
  }
}

extern "C" void kernel_launch(void* const* d_in, const int* in_sizes, int n_in,
                              void* d_out, int out_size, void* d_ws, size_t ws_size,
                              hipStream_t stream) {
  (void)in_sizes; (void)n_in; (void)out_size; (void)ws_size;
  const float* D = (const float*)d_in[0];       // (4, 2048, 2048) f32
  // d_in[1] is K == 8 (compile-time constant here)

  float* C     = (float*)d_out;                             // (4,2048,8)
  float* Call  = C + (size_t)BATCH * NDIM * KSTEPS;         // (4,2048,8,2048)
  float* Cbuf0 = (float*)d_ws;                              // DP carry ping
  float* Cbuf1 = Cbuf0 + BATCH * NDIM;                      // DP carry pong

  dim3 grid(BATCH * NDIM), block(TPB);
  ctab_step0<<<grid, block, 0, stream>>>(D, C, Call, Cbuf0);
  for (int kk = 1; kk < KSTEPS; ++kk) {
    const float* Cprev = (kk & 1) ? Cbuf0 : Cbuf1;
    float*       Ccur  = (kk & 1) ? Cbuf1 : Cbuf0;
    ctab_step<<<grid, block, 0, stream>>>(D, Cprev, Ccur, C, Call, kk);
  }
}